// SO2TC_33414845563121
// MI455X (gfx1250) — compile-verified
//
#include <hip/hip_runtime.h>

// SO2 tensor-contraction, algebraically reduced:
//   out1[b,c] = T2[b,c,0,0] * outer( M1[b] @ (T1[b,c] @ g1[b]), s0[b] )
//   out2[b,c] = T2[b,c,0,0] * outer( M2[b] @ (T1[b,c] @ g2[b]), s0[b] )
// with R2 = R@R, M1 = NORM*S@R, M2 = NORM*S@R2, g1 = R[0,:], g2 = R2[0,:], s0 = S[:,0].
//
// Memory-bound (~605 MB/call, <0.5 GFLOP): stream T1 tiles into LDS via the
// CDNA5 async global->LDS path, compute from LDS, non-temporal B128 stores.

#define B_TOT 65536
#define C_TOT 64
#define PAIRS (B_TOT * C_TOT)          // 4,194,304
#define PAIRS_PER_BLOCK 1024           // 16 b's * 64 c's
#define B_PER_BLOCK 16
#define THREADS 256
#define GRID (PAIRS / PAIRS_PER_BLOCK) // 4096

typedef int   v4i __attribute__((ext_vector_type(4)));
typedef float v4f __attribute__((ext_vector_type(4)));
typedef __attribute__((address_space(1))) v4i* gptr_v4i;
typedef __attribute__((address_space(3))) v4i* lptr_v4i;

// ---- CDNA5 async global->LDS copy (B128) ------------------------------------
__device__ __forceinline__ void async_ld_b128(const float* g, float* l) {
#if __has_builtin(__builtin_amdgcn_global_load_async_to_lds_b128)
  __builtin_amdgcn_global_load_async_to_lds_b128(
      (gptr_v4i)g, (lptr_v4i)l, /*imm offset*/ 0, /*cpol*/ 0);
#else
  unsigned lds_off = (unsigned)(unsigned long long)l;   // low 32 bits = LDS offset
  unsigned long long ga = (unsigned long long)g;
  asm volatile("global_load_async_to_lds_b128 %0, %1, off"
               :: "v"(lds_off), "v"(ga) : "memory");
#endif
}

__device__ __forceinline__ void wait_async0() {
#if __has_builtin(__builtin_amdgcn_s_wait_asynccnt)
  __builtin_amdgcn_s_wait_asynccnt(0);
#else
  asm volatile("s_wait_asynccnt 0" ::: "memory");
#endif
}
// -----------------------------------------------------------------------------

__global__ __launch_bounds__(THREADS)
void SO2TC_kernel(const float* __restrict__ T1g, const float* __restrict__ T2g,
                  const float* __restrict__ Rg,  const float* __restrict__ Sg,
                  float* __restrict__ out) {
  __shared__ __align__(16) float sT1[PAIRS_PER_BLOCK * 9];  // 36 KB contiguous T1 tile
  __shared__ float sC[B_PER_BLOCK][28];                     // per-b coefficients

  const int t = threadIdx.x;
  const int blk = blockIdx.x;
  const size_t base_pair = (size_t)blk * PAIRS_PER_BLOCK;

  // ---- stage contiguous T1 tile (9216 floats = 2304 x B128) asynchronously --
  {
    const float* gbase = T1g + base_pair * 9;   // 16B-aligned (1024*9*4 % 16 == 0)
#pragma unroll
    for (int j = 0; j < 9; ++j) {
      const int chunk = t + j * THREADS;        // contiguous per wave-instruction
      async_ld_b128(gbase + chunk * 4, &sT1[chunk * 4]);
    }
  }

  // ---- per-b coefficient setup (threads 0..15, one b each) -------------------
  if (t < B_PER_BLOCK) {
    const int b = blk * B_PER_BLOCK + t;
    float R[9], S[9];
#pragma unroll
    for (int i = 0; i < 9; ++i) { R[i] = Rg[(size_t)b * 9 + i]; S[i] = Sg[(size_t)b * 9 + i]; }
    float R2[9];
#pragma unroll
    for (int i = 0; i < 3; ++i)
#pragma unroll
      for (int k = 0; k < 3; ++k)
        R2[i*3+k] = R[i*3+0]*R[0*3+k] + R[i*3+1]*R[1*3+k] + R[i*3+2]*R[2*3+k];
    const float NORM = 0.57735026918962576451f;   // 1/sqrt(3^K), K=1
#pragma unroll
    for (int i = 0; i < 3; ++i)
#pragma unroll
      for (int m = 0; m < 3; ++m) {
        sC[t][i*3+m]      = NORM*(S[i*3+0]*R[0*3+m]  + S[i*3+1]*R[1*3+m]  + S[i*3+2]*R[2*3+m]);
        sC[t][12+i*3+m]   = NORM*(S[i*3+0]*R2[0*3+m] + S[i*3+1]*R2[1*3+m] + S[i*3+2]*R2[2*3+m]);
      }
#pragma unroll
    for (int n = 0; n < 3; ++n) {
      sC[t][9 + n]  = R[n];        // g1 = R[0,:]
      sC[t][21 + n] = R2[n];       // g2 = R2[0,:]
      sC[t][24 + n] = S[n * 3];    // s0 = S[:,0]
    }
  }

  wait_async0();       // own async B128s done
  __syncthreads();     // all waves' async writes + coeff writes visible

  // ---- compute: each thread owns 4 consecutive pairs of one b ----------------
  const int bl = t >> 4;                 // local b 0..15
  const int ci = (t & 15) << 2;          // c base 0..60
  const int pair_local = bl * C_TOT + ci;
  const size_t gp0 = base_pair + pair_local;

  float M1[9], M2[9], g1[3], g2[3], s0[3];
#pragma unroll
  for (int i = 0; i < 9; ++i) { M1[i] = sC[bl][i]; M2[i] = sC[bl][12 + i]; }
#pragma unroll
  for (int n = 0; n < 3; ++n) { g1[n] = sC[bl][9+n]; g2[n] = sC[bl][21+n]; s0[n] = sC[bl][24+n]; }

  __align__(16) float o1[36];
  __align__(16) float o2[36];

#pragma unroll
  for (int q = 0; q < 4; ++q) {
    const float* T = &sT1[(pair_local + q) * 9];
    // T2[b,c,0,0]: strided read-once gather -> non-temporal load
    const float t2s = __builtin_nontemporal_load(T2g + (gp0 + q) * 9);
    float u1[3], u2[3];
#pragma unroll
    for (int i = 0; i < 3; ++i) {
      u1[i] = T[i*3+0]*g1[0] + T[i*3+1]*g1[1] + T[i*3+2]*g1[2];
      u2[i] = T[i*3+0]*g2[0] + T[i*3+1]*g2[1] + T[i*3+2]*g2[2];
    }
    float ss[3];
#pragma unroll
    for (int k = 0; k < 3; ++k) ss[k] = s0[k] * t2s;
#pragma unroll
    for (int i = 0; i < 3; ++i) {
      const float a1 = M1[i*3+0]*u1[0] + M1[i*3+1]*u1[1] + M1[i*3+2]*u1[2];
      const float a2 = M2[i*3+0]*u2[0] + M2[i*3+1]*u2[1] + M2[i*3+2]*u2[2];
#pragma unroll
      for (int k = 0; k < 3; ++k) {
        o1[q*9 + i*3 + k] = a1 * ss[k];
        o2[q*9 + i*3 + k] = a2 * ss[k];
      }
    }
  }

  // ---- 16B-aligned coalesced non-temporal B128 stores (144 B per out/thread) -
  v4f* p1 = (v4f*)(out + gp0 * 9);
  v4f* p2 = (v4f*)(out + (size_t)PAIRS * 9 + gp0 * 9);
#pragma unroll
  for (int j = 0; j < 9; ++j) {
    __builtin_nontemporal_store(((const v4f*)o1)[j], p1 + j);
    __builtin_nontemporal_store(((const v4f*)o2)[j], p2 + j);
  }
}

extern "C" void kernel_launch(void* const* d_in, const int* in_sizes, int n_in,
                              void* d_out, int out_size, void* d_ws, size_t ws_size,
                              hipStream_t stream) {
  const float* T1 = (const float*)d_in[0];
  const float* T2 = (const float*)d_in[1];
  const float* R  = (const float*)d_in[2];
  const float* S  = (const float*)d_in[3];
  (void)in_sizes; (void)n_in; (void)out_size; (void)d_ws; (void)ws_size;
  SO2TC_kernel<<<GRID, THREADS, 0, stream>>>(T1, T2, R, S, (float*)d_out);
}